// MedialSurface_1365799600955
// MI455X (gfx1250) — compile-verified
//
#include <hip/hip_runtime.h>
#include <stdint.h>

// Medial surface: per-axis Zhang-Suen thinning to convergence, OR, 6-dilate, AND mask.
// Volume: [1, D=48, H=384, W=384] float32 of {0,1}.
// Strategy: bit-pack 32 voxels/word along the 384 axis; each slice thins entirely
// in LDS (<=18KB) to local convergence; wave32 ballot does pack/transpose.

#define DD 48
#define HH 384
#define WW 384
#define CW 12                       // 384 / 32 words per packed row
#define SLICE_WORDS_MAX (HH * CW)   // 4608 words (z-pass slice)

typedef uint32_t u32;

#if __has_builtin(__builtin_amdgcn_global_load_async_to_lds_b32) && \
    __has_builtin(__builtin_amdgcn_s_wait_asynccnt)
#define USE_ASYNC_LDS 1
#else
#define USE_ASYNC_LDS 0
#endif

__device__ __forceinline__ u32 ballot32(bool p) {
#if __has_builtin(__builtin_amdgcn_ballot_w32)
  return __builtin_amdgcn_ballot_w32(p);
#else
  return (u32)__ballot((int)p);
#endif
}

// ---------------------------------------------------------------------------
// Packing: one wave32 builds one 32-bit word via ballot (lane i <-> bit i).
// pack_xy also emits the y-pass permuted copy (same bit axis = x).
// ---------------------------------------------------------------------------
__global__ __launch_bounds__(256) void pack_xy_kernel(const float* __restrict__ vol,
                                                      u32* __restrict__ mask_c,
                                                      u32* __restrict__ zbuf,
                                                      u32* __restrict__ ybuf) {
  const uint32_t t = blockIdx.x * 256u + threadIdx.x;   // voxel id, x fastest
  const float v = vol[t];
  const u32 word = ballot32(v == 1.0f);
  if ((threadIdx.x & 31u) == 0u) {
    const uint32_t x  = t % WW;
    const uint32_t y  = (t / WW) % HH;
    const uint32_t z  = t / (WW * HH);
    const uint32_t wc = x >> 5;
    const uint32_t iz = (z * HH + y) * CW + wc;         // canonical (z,y | x bits)
    const uint32_t iy = (y * DD + z) * CW + wc;         // y-pass   (y,d | x bits)
    mask_c[iz] = word;
    zbuf[iz]   = word;
    ybuf[iy]   = word;
  }
}

// x-pass layout: slice per x, rows d, bits along y (bit transpose via ballot).
__global__ __launch_bounds__(256) void pack_x_kernel(const float* __restrict__ vol,
                                                     u32* __restrict__ xbuf) {
  const uint32_t t    = blockIdx.x * 256u + threadIdx.x;
  const uint32_t wid  = t >> 5;            // word id: ((x*DD)+d)*CW + wcy
  const uint32_t lane = t & 31u;
  const uint32_t wcy  = wid % CW;
  const uint32_t d    = (wid / CW) % DD;
  const uint32_t x    = wid / (CW * DD);
  const uint32_t y    = wcy * 32u + lane;
  const float v = vol[(d * HH + y) * WW + x];
  const u32 word = ballot32(v == 1.0f);
  if (lane == 0u) xbuf[wid] = word;
}

// ---------------------------------------------------------------------------
// Bit-sliced Zhang-Suen sub-iteration on an R x 384 slice packed in LDS.
// ---------------------------------------------------------------------------
__device__ __forceinline__ bool zs_subiter(const u32* __restrict__ src,
                                           u32* __restrict__ dst,
                                           int R, bool first, int tid) {
  bool changed = false;
  const int words = R * CW;
  for (int idx = tid; idx < words; idx += 256) {
    const int r  = idx / CW;
    const int wc = idx - r * CW;
    const u32 cur = src[idx];
    const u32 uc = (r > 0)                    ? src[idx - CW]     : 0u;
    const u32 ul = (r > 0 && wc > 0)          ? src[idx - CW - 1] : 0u;
    const u32 ur = (r > 0 && wc < CW - 1)     ? src[idx - CW + 1] : 0u;
    const u32 ml = (wc > 0)                   ? src[idx - 1]      : 0u;
    const u32 mr = (wc < CW - 1)              ? src[idx + 1]      : 0u;
    const u32 dc = (r < R - 1)                ? src[idx + CW]     : 0u;
    const u32 dl = (r < R - 1 && wc > 0)      ? src[idx + CW - 1] : 0u;
    const u32 dr = (r < R - 1 && wc < CW - 1) ? src[idx + CW + 1] : 0u;

    // P2..P9 clockwise from north; bit i <-> column wc*32+i.
    const u32 P2 = uc;
    const u32 P3 = (uc  >> 1) | (ur << 31);   // NE (east = col+1 = bit i+1)
    const u32 P4 = (cur >> 1) | (mr << 31);   // E
    const u32 P5 = (dc  >> 1) | (dr << 31);   // SE
    const u32 P6 = dc;                        // S
    const u32 P7 = (dc  << 1) | (dl >> 31);   // SW
    const u32 P8 = (cur << 1) | (ml >> 31);   // W
    const u32 P9 = (uc  << 1) | (ul >> 31);   // NW

    // B = popcount of 8 neighbors, bit-sliced carry-save adders -> 4-bit count.
    const u32 sa = P2 ^ P3, ca = P2 & P3;
    const u32 sb = P4 ^ P5, cb = P4 & P5;
    const u32 sc = P6 ^ P7, cc = P6 & P7;
    const u32 sd = P8 ^ P9, cd = P8 & P9;
    const u32 ab0 = sa ^ sb, kab = sa & sb;
    const u32 tx1 = ca ^ cb;
    const u32 ab1 = tx1 ^ kab;
    const u32 ab2 = (ca & cb) | (kab & tx1);
    const u32 cd0 = sc ^ sd, kcd = sc & sd;
    const u32 tx2 = cc ^ cd;
    const u32 cd1 = tx2 ^ kcd;
    const u32 cd2 = (cc & cd) | (kcd & tx2);
    const u32 b0 = ab0 ^ cd0, k0 = ab0 & cd0;
    const u32 tx3 = ab1 ^ cd1;
    const u32 b1 = tx3 ^ k0;
    const u32 k1 = (ab1 & cd1) | (k0 & tx3);
    const u32 tx4 = ab2 ^ cd2;
    const u32 b2 = tx4 ^ k1;
    const u32 b3 = (ab2 & cd2) | (k1 & tx4);
    // 2 <= B <= 6
    const u32 condB = (b1 | b2) & ~b3 & ~(b0 & b1 & b2);

    // A == 1: exactly one 0->1 transition around ring P2..P9..P2.
    u32 acc = 0u, multi = 0u, tk;
    tk = ~P2 & P3; multi |= acc & tk; acc |= tk;
    tk = ~P3 & P4; multi |= acc & tk; acc |= tk;
    tk = ~P4 & P5; multi |= acc & tk; acc |= tk;
    tk = ~P5 & P6; multi |= acc & tk; acc |= tk;
    tk = ~P6 & P7; multi |= acc & tk; acc |= tk;
    tk = ~P7 & P8; multi |= acc & tk; acc |= tk;
    tk = ~P8 & P9; multi |= acc & tk; acc |= tk;
    tk = ~P9 & P2; multi |= acc & tk; acc |= tk;
    const u32 condA = acc & ~multi;

    const u32 condC = first ? (~(P2 & P4 & P6) & ~(P4 & P6 & P8))
                            : (~(P2 & P4 & P8) & ~(P2 & P6 & P8));

    const u32 rem = cur & condB & condA & condC;
    const u32 out = cur & ~rem;
    dst[idx] = out;
    changed |= (out != cur);
  }
  return changed;
}

// One workgroup per slice; iterate in LDS (double-buffered) to local convergence.
// blocks [0,48): z-pass R=384; [48,432): y-pass R=48; [432,816): x-pass R=48.
__global__ __launch_bounds__(256) void thin_all_kernel(u32* __restrict__ zbuf,
                                                       u32* __restrict__ ybuf,
                                                       u32* __restrict__ xbuf) {
  __shared__ u32 Abuf[SLICE_WORDS_MAX];
  __shared__ u32 Bbuf[SLICE_WORDS_MAX];
  __shared__ int s_flag;

  u32* gsrc;
  int R;
  const int b = blockIdx.x;
  if (b < DD)            { gsrc = zbuf + b * (HH * CW);              R = HH; }
  else if (b < DD + HH)  { gsrc = ybuf + (b - DD) * (DD * CW);       R = DD; }
  else                   { gsrc = xbuf + (b - DD - HH) * (DD * CW);  R = DD; }

  const int tid   = threadIdx.x;
  const int words = R * CW;

  // Stage slice into LDS (async global->LDS where the toolchain exposes it).
  for (int idx = tid; idx < words; idx += 256) {
#if USE_ASYNC_LDS
    __builtin_amdgcn_global_load_async_to_lds_b32(
        (__attribute__((address_space(1))) int*)(gsrc + idx),
        (__attribute__((address_space(3))) int*)(&Abuf[idx]),
        /*imm offset*/ 0, /*cpol*/ 0);
#else
    Abuf[idx] = gsrc[idx];
#endif
  }
#if USE_ASYNC_LDS
  __builtin_amdgcn_s_wait_asynccnt(0);
#endif
  __syncthreads();

  for (int it = 0; it < 512; ++it) {          // safety cap; converges far earlier
    if (tid == 0) s_flag = 0;
    __syncthreads();
    bool ch = zs_subiter(Abuf, Bbuf, R, /*first=*/true, tid);
    __syncthreads();
    ch |= zs_subiter(Bbuf, Abuf, R, /*first=*/false, tid);
    if (ch) s_flag = 1;
    __syncthreads();
    if (s_flag == 0) break;                   // uniform: thinning is monotone
    __syncthreads();
  }

  for (int idx = tid; idx < words; idx += 256) gsrc[idx] = Abuf[idx];
}

// OR the three skeletons into canonical layout (in place over zbuf).
// x-pass bits run along y -> re-transpose with wave32 ballot.
__global__ __launch_bounds__(256) void combine_kernel(u32* __restrict__ zb,
                                                      const u32* __restrict__ yb,
                                                      const u32* __restrict__ xb) {
  const uint32_t t    = blockIdx.x * 256u + threadIdx.x;   // canonical voxel id
  const uint32_t lane = threadIdx.x & 31u;
  const uint32_t x = t % WW;
  const uint32_t y = (t / WW) % HH;
  const uint32_t z = t / (WW * HH);
  const u32* xp = &xb[(x * DD + z) * CW + (y >> 5)];
  __builtin_prefetch(xp, 0, 1);
  const u32 bit = (*xp >> (y & 31u)) & 1u;
  const u32 xword = ballot32(bit != 0u);
  if (lane == 0u) {
    const uint32_t wc  = x >> 5;
    const uint32_t idx = (z * HH + y) * CW + wc;
    zb[idx] = zb[idx] | yb[(y * DD + z) * CW + wc] | xword;   // 1:1 -> safe in place
  }
}

// Fused: 6-connected dilation (border 0) + AND mask + unpack to float32.
__global__ __launch_bounds__(256) void finalize_kernel(const u32* __restrict__ skel,
                                                       const u32* __restrict__ mask,
                                                       float* __restrict__ out) {
  const uint32_t idx = blockIdx.x * 256u + threadIdx.x;     // canonical word id
  const uint32_t wc = idx % CW;
  const uint32_t y  = (idx / CW) % HH;
  const uint32_t z  = idx / (CW * HH);
  const u32 c  = skel[idx];
  const u32 zp = (z > 0)      ? skel[idx - HH * CW] : 0u;
  const u32 zn = (z < DD - 1) ? skel[idx + HH * CW] : 0u;
  const u32 yp = (y > 0)      ? skel[idx - CW] : 0u;
  const u32 yn = (y < HH - 1) ? skel[idx + CW] : 0u;
  const u32 xl = (wc > 0)      ? skel[idx - 1] : 0u;
  const u32 xr = (wc < CW - 1) ? skel[idx + 1] : 0u;
  u32 d6 = c | zp | zn | yp | yn
             | ((c << 1) | (xl >> 31))    // west neighbor contribution
             | ((c >> 1) | (xr << 31));   // east neighbor contribution
  d6 &= mask[idx];
  const uint32_t base = (z * HH + y) * WW + wc * 32u;
  float4* o4 = (float4*)(out + base);
#pragma unroll
  for (int q = 0; q < 8; ++q) {
    float4 f;
    f.x = (d6 >> (4 * q + 0)) & 1u ? 1.0f : 0.0f;
    f.y = (d6 >> (4 * q + 1)) & 1u ? 1.0f : 0.0f;
    f.z = (d6 >> (4 * q + 2)) & 1u ? 1.0f : 0.0f;
    f.w = (d6 >> (4 * q + 3)) & 1u ? 1.0f : 0.0f;
    o4[q] = f;
  }
}

extern "C" void kernel_launch(void* const* d_in, const int* in_sizes, int n_in,
                              void* d_out, int out_size, void* d_ws, size_t ws_size,
                              hipStream_t stream) {
  (void)in_sizes; (void)n_in; (void)out_size; (void)ws_size;
  const float* vol = (const float*)d_in[0];
  float* out = (float*)d_out;

  const size_t WORDS = (size_t)DD * HH * CW;   // 221,184 words = 884,736 B / buffer
  u32* mask_c = (u32*)d_ws;                    // canonical packed mask (kept)
  u32* zbuf   = mask_c + WORDS;                // z-pass buffer -> combined skeleton
  u32* ybuf   = zbuf + WORDS;                  // y-pass buffer
  u32* xbuf   = ybuf + WORDS;                  // x-pass buffer (bits along y)

  const int vox = DD * HH * WW;                // 7,077,888 (multiple of 256)

  pack_xy_kernel <<<vox / 256, 256, 0, stream>>>(vol, mask_c, zbuf, ybuf);
  pack_x_kernel  <<<vox / 256, 256, 0, stream>>>(vol, xbuf);
  thin_all_kernel<<<DD + HH + WW, 256, 0, stream>>>(zbuf, ybuf, xbuf);
  combine_kernel <<<vox / 256, 256, 0, stream>>>(zbuf, ybuf, xbuf);
  finalize_kernel<<<(int)(WORDS / 256), 256, 0, stream>>>(zbuf, mask_c, out);
}